// Loss_45887430590701
// MI455X (gfx1250) — compile-verified
//
#include <hip/hip_runtime.h>
#include <math.h>

// CDNA5 / gfx1250, wave32. Tree-scan loss reformulated in linear space:
// prev_new = (E x T) * invC via V_WMMA_F32_16X16X4_F32 (full f32).
// The normalizer C = E.u is folded into the matmul as T column 31 (unused,
// since P = 31 < 32), so it falls out of the same WMMA chain as D column 31.
// T fragments are hoisted into registers (32 VGPRs); no reduction-mask
// registers -> no scratch spills.

typedef __attribute__((ext_vector_type(2))) float v2f;
typedef __attribute__((ext_vector_type(8))) float v8f;

#define PP 32          // padded P (P = 2L-1 = 31 for DEPTH=4)
#define ROWPAD 36      // row stride: 16B-aligned 16-col halves, conflict-free A reads
#define EPS_F 1.6155871338926322e-28f   // exp(-64)

// ---------------------------------------------------------------------------
// Kernel 1: latency-bound random gather. W[t][b][p] = exp(log_probs[p,b,targets[t,b]])
// One thread per element -> 64K independent loads in flight. Pad col 31 = 0
// (this also keeps E column 31 zero, protecting the folded C-column in T).
// ---------------------------------------------------------------------------
__global__ void gather_exp_kernel(const float* __restrict__ log_probs,
                                  const int* __restrict__ targets,
                                  float* __restrict__ W,
                                  int L, int B, long long V, int P) {
  int tid = blockIdx.x * blockDim.x + threadIdx.x;
  int total = L * B * PP;
  if (tid >= total) return;
  int p = tid & (PP - 1);
  int b = (tid / PP) % B;
  int t = tid / (PP * B);
  float val = 0.0f;
  if (p < P) {
    int tgt = targets[t * B + b];
    float lp = log_probs[((long long)p * B + b) * V + tgt];
    val = expf(lp);
  }
  W[tid] = val;
}

// ---------------------------------------------------------------------------
// Kernel 2: the scan. One wave (32 lanes) owns 16 batch rows for all L steps.
// Fragment layouts per ISA 7.12.2:
//   A 16x4 : lane -> M=lane%16 ; VGPR j holds K = 4k + 2*(lane>=16) + j
//   B 4x16 : lane -> N=lane%16 ; VGPR j holds K = 4k + 2*(lane>=16) + j
//   D 16x16: VGPR r, lane l -> M = r + 8*(l>=16), N = l%16
// D column 31 (lanes 15/31, tile H) carries C[row] = sum_q E[row,q]*u[q].
// ---------------------------------------------------------------------------
__global__ void tree_scan_kernel(const float* __restrict__ W,         // may be null
                                 const float* __restrict__ log_probs, // fallback gather
                                 const int* __restrict__ targets,
                                 const int* __restrict__ target_lengths,
                                 const int* __restrict__ out_uniq_idx, int n_uniq,
                                 const int* __restrict__ out_uniq_inv,
                                 const int* __restrict__ in_idx, int n_adj,
                                 const int* __restrict__ start_idxs, int n_start,
                                 const int* __restrict__ end_idxs, int n_end,
                                 float* __restrict__ out,
                                 int L, int B, long long V, int P) {
  __shared__ float Tm[PP][PP];           // T[q][p] (0/1); col 31 = u[q]
  __shared__ float uvec[PP];             // indicator of out_uniq nodes
  __shared__ float evec[PP];             // indicator of end nodes
  __shared__ __align__(16) float Esh[16][ROWPAD];    // E tile
  __shared__ __align__(16) float prevSh[16][ROWPAD]; // prev state tile
  __shared__ float Cb[16];               // C per batch row (from WMMA col 31)
  __shared__ float accSh[16];
  __shared__ int   tlSh[16];

  const int l  = threadIdx.x;   // 0..31 (one wave)
  const int hi = l >> 4;        // lane half
  const int lm = l & 15;
  const int b0 = blockIdx.x * 16;
  const int brow = b0 + lm;     // this lane's batch row

  // ---- build constants in LDS ----
  for (int i = l; i < PP * PP; i += 32) (&Tm[0][0])[i] = 0.0f;
  if (l < PP) { uvec[l] = 0.0f; evec[l] = 0.0f; }
  for (int j = 0; j < 16; ++j) prevSh[lm][hi * 16 + j] = 0.0f;
  if (l < 16) {
    accSh[l] = 0.0f;
    tlSh[l]  = (b0 + l < B) ? target_lengths[b0 + l] : 0;
  }
  __syncthreads();
  for (int a = l; a < n_adj; a += 32) {
    int q = out_uniq_idx[out_uniq_inv[a]];   // out_idx[a]
    int p = in_idx[a];                       // in_idx < P <= 31, never col 31
    atomicAdd(&Tm[q][p], 1.0f);
  }
  for (int i = l; i < n_uniq; i += 32) uvec[out_uniq_idx[i]] = 1.0f;
  for (int i = l; i < n_end;  i += 32) evec[end_idxs[i]]     = 1.0f;
  if (l < 16)
    for (int i = 0; i < n_start; ++i) prevSh[l][start_idxs[i]] = 1.0f;
  __syncthreads();
  if (l < PP) Tm[l][PP - 1] = uvec[l];   // fold normalizer column into T
  __syncthreads();

  // ---- hoist T fragments into registers (constant over t) ----
  v2f tL[8], tH[8];
#pragma unroll
  for (int k = 0; k < 8; ++k) {
    int kr = 4 * k + 2 * hi;
    tL[k].x = Tm[kr][lm];      tL[k].y = Tm[kr + 1][lm];
    tH[k].x = Tm[kr][16 + lm]; tH[k].y = Tm[kr + 1][16 + lm];
  }

  // ---- sequential scan over t ----
  for (int t = 0; t < L; ++t) {
    // Stage 1: E = max(prev, EPS) * exp(lp_t)  (== exp(log_safe(prev) + lp_t))
    if (W != nullptr) {
      const float4* Wr4 =
          (const float4*)(W + (((long long)t * B + brow) * PP + hi * 16));
      float4*       Erow = (float4*)&Esh[lm][hi * 16];
      const float4* Prow = (const float4*)&prevSh[lm][hi * 16];
#pragma unroll
      for (int q = 0; q < 4; ++q) {
        float4 w  = (brow < B) ? Wr4[q] : make_float4(0.f, 0.f, 0.f, 0.f);
        float4 pv = Prow[q];
        float4 E;
        E.x = fmaxf(pv.x, EPS_F) * w.x;
        E.y = fmaxf(pv.y, EPS_F) * w.y;
        E.z = fmaxf(pv.z, EPS_F) * w.z;
        E.w = fmaxf(pv.w, EPS_F) * w.w;
        Erow[q] = E;
      }
    } else {
      int tgt = (brow < B) ? targets[t * B + brow] : 0;
#pragma unroll
      for (int j = 0; j < 16; ++j) {
        int p = hi * 16 + j;
        float w = 0.0f;
        if (brow < B && p < P)
          w = expf(log_probs[((long long)p * B + brow) * V + tgt]);
        Esh[lm][p] = fmaxf(prevSh[lm][p], EPS_F) * w;
      }
    }
    __syncthreads();

    // Stage 2: D = E x [T | u] via V_WMMA_F32_16X16X4_F32 (EXEC all-1s,
    // B operands from registers). d1 column 31 (lanes 15/31) = C[row].
    v8f d0 = {0.f,0.f,0.f,0.f,0.f,0.f,0.f,0.f};
    v8f d1 = {0.f,0.f,0.f,0.f,0.f,0.f,0.f,0.f};
#pragma unroll
    for (int k = 0; k < 8; ++k) {
      int kr = 4 * k + 2 * hi;
      v2f a; a.x = Esh[lm][kr]; a.y = Esh[lm][kr + 1];
      // 8 args: (neg_a, A, neg_b, B, c_mod, C, reuse_a, reuse_b)
      d0 = __builtin_amdgcn_wmma_f32_16x16x4_f32(false, a, false, tL[k], (short)0, d0, false, false);
      d1 = __builtin_amdgcn_wmma_f32_16x16x4_f32(false, a, false, tH[k], (short)0, d1, false, false);
    }
    if (lm == 15) {                        // N=31 holders publish C
#pragma unroll
      for (int r = 0; r < 8; ++r) Cb[r + 8 * hi] = d1[r];
    }
    __syncthreads();

    // Stage 3: loss accumulation + normalize/write prev_new.
    if (l < 16) {
      int tl = tlSh[l];
      if (t + 1 == tl) {                   // rare: one t per row over the scan
        float e = 0.0f;
        for (int p = 0; p < PP; ++p) e += Esh[l][p] * evec[p];
        accSh[l] += logf(e);               // end_vals
      }
      if ((t + 1 < tl) && (t + 1 < L)) accSh[l] += logf(Cb[l]);  // log_C
    }
#pragma unroll
    for (int r = 0; r < 8; ++r) {
      int row = r + 8 * hi;                // D-matrix M index (ISA 7.12.2)
      float ic = __builtin_amdgcn_rcpf(Cb[row]);
      prevSh[row][lm]      = d0[r] * ic;   // N tile 0
      prevSh[row][16 + lm] = d1[r] * ic;   // N tile 1 (col 31 -> ~1, harmless:
                                           //   E col 31 is forced 0 by w=0)
    }
    __syncthreads();
  }

  if (l < 16 && b0 + l < B) out[b0 + l] = -accSh[l];
}

// ---------------------------------------------------------------------------
extern "C" void kernel_launch(void* const* d_in, const int* in_sizes, int n_in,
                              void* d_out, int out_size, void* d_ws, size_t ws_size,
                              hipStream_t stream) {
  const float* log_probs      = (const float*)d_in[0];
  const int*   targets        = (const int*)d_in[1];
  const int*   target_lengths = (const int*)d_in[2];
  const int*   out_uniq_idx   = (const int*)d_in[3];
  const int*   out_uniq_inv   = (const int*)d_in[4];
  const int*   in_idx         = (const int*)d_in[5];
  const int*   start_idxs     = (const int*)d_in[6];
  const int*   end_idxs       = (const int*)d_in[7];

  const int B = in_sizes[2];                 // target_lengths is (B,)
  const int L = in_sizes[1] / B;             // targets is (L,B)
  const int P = 2 * L - 1;                   // full binary tree: P = 2^(DEPTH+1)-1
  const long long V = (long long)in_sizes[0] / ((long long)P * B);
  const int n_uniq  = in_sizes[3];
  const int n_adj   = in_sizes[5];
  const int n_start = in_sizes[6];
  const int n_end   = in_sizes[7];

  // Phase 1: gather+exp into workspace (maximum memory-level parallelism).
  float* W = nullptr;
  size_t wbytes = (size_t)L * (size_t)B * PP * sizeof(float);
  if (ws_size >= wbytes) {
    W = (float*)d_ws;
    int total = L * B * PP;
    gather_exp_kernel<<<(total + 255) / 256, 256, 0, stream>>>(
        log_probs, targets, W, L, B, V, P);
  }

  // Phase 2: scan. One wave per 16 batch rows.
  tree_scan_kernel<<<(B + 15) / 16, 32, 0, stream>>>(
      W, log_probs, targets, target_lengths,
      out_uniq_idx, n_uniq, out_uniq_inv,
      in_idx, n_adj, start_idxs, n_start, end_idxs, n_end,
      (float*)d_out, L, B, V, P);
}